// TF_NP_MCCD_OPD_v2_53137335386170
// MI455X (gfx1250) — compile-verified
//
#include <hip/hip_runtime.h>
#include <hip/hip_bf16.h>
#include <climits>

typedef __attribute__((ext_vector_type(2))) float v2f;
typedef __attribute__((ext_vector_type(8))) float v8f;

#define N_STARS 2048
#define BATCH   256
#define PF      28
#define NGE     512
#define GF      32
#define OPD2    65536          // 256*256 flattened OPD pixels
#define KPAD    64             // padded K (28 poly + 32 graph + 4 pad)

// ---------------------------------------------------------------------------
// Kernel 1: index search (faithful first-True flat index over (N,2) equality)
// plus the two small projections, producing interm[b][0..59] (padded to 64):
//   cols 0..27  = poly_dic[idx] @ alpha_poly
//   cols 28..59 = graph_dic[idx] @ alpha_graph
// One block (64 threads) per batch element.
// ---------------------------------------------------------------------------
__global__ __launch_bounds__(64)
void build_interm_kernel(const float* __restrict__ positions,
                         const float* __restrict__ obs_pos,
                         const float* __restrict__ poly_dic,
                         const float* __restrict__ graph_dic,
                         const float* __restrict__ alpha_poly,
                         const float* __restrict__ alpha_graph,
                         float* __restrict__ interm) {
    __shared__ int s_min;
    const int b = blockIdx.x;
    const int t = threadIdx.x;          // 0..63
    if (t == 0) s_min = INT_MAX;
    __syncthreads();

    const float px = positions[2 * b];
    const float py = positions[2 * b + 1];
    int best = INT_MAX;
    for (int s = t; s < N_STARS; s += 64) {
        const float ox = obs_pos[2 * s];
        const float oy = obs_pos[2 * s + 1];
        if (ox == px)       best = min(best, 2 * s);      // eq[s,0] first in row-major
        else if (oy == py)  best = min(best, 2 * s + 1);  // eq[s,1]
    }
    atomicMin(&s_min, best);
    __syncthreads();
    int flat = s_min;
    if (flat == INT_MAX) flat = 0;      // argmax of all-False -> 0 (matches jnp.argmax)
    const int idx = flat >> 1;          // row index

    float acc = 0.0f;
    if (t < PF) {
        const float* pd = poly_dic + (size_t)idx * PF;
        #pragma unroll
        for (int p = 0; p < PF; ++p)
            acc += pd[p] * alpha_poly[p * PF + t];
    } else if (t < PF + GF) {           // 28..59
        const int j = t - PF;
        const float* gd = graph_dic + (size_t)idx * NGE;
        for (int e = 0; e < NGE; ++e)
            acc += gd[e] * alpha_graph[e * GF + j];
    }                                   // t 60..63 -> 0 (padding, never read)
    interm[(size_t)b * KPAD + t] = acc;
}

// ---------------------------------------------------------------------------
// Kernel 2: out(256 x 65536) = interm(256 x 60) @ [S_poly; S_graph](60 x 65536)
// via V_WMMA_F32_16X16X4_F32. One wave per 16-column N-tile; B fragments for
// all 15 K-steps are loaded once and reused across all 16 M-tiles.
//
// Layouts (ISA 7.12.2, wave32):
//  A 16x4 f32 (2 VGPR):  lane&15 = M; lane-half selects K pair (0,1)/(2,3)
//  B 4x16 f32 (2 VGPR):  lane&15 = N; lane-half selects K pair (0,1)/(2,3)
//  C 16x16 f32 (8 VGPR): VGPR j -> M = j + 8*(lane>=16); lane&15 = N
// ---------------------------------------------------------------------------
__global__ __launch_bounds__(256)
void opd_wmma_gemm_kernel(const float* __restrict__ interm,
                          const float* __restrict__ S_poly,
                          const float* __restrict__ S_graph,
                          float* __restrict__ out) {
    const int lane  = threadIdx.x & 31;
    const int wave  = threadIdx.x >> 5;
    const int ntile = blockIdx.x * 8 + wave;        // 0..4095
    const int n     = ntile * 16 + (lane & 15);     // output column
    const int koff  = (lane >> 4) * 2;              // 0 for lanes 0-15, 2 for 16-31

    // Preload B fragments for the 15 K-steps (K = 0..59 of [S_poly; S_graph]).
    // 28 % 4 == 0, so each step lies entirely in S_poly (k0<28) or S_graph.
    v2f bfrag[15];
    #pragma unroll
    for (int s = 0; s < 15; ++s) {
        const int k0 = 4 * s;
        const int r  = k0 + koff;
        const float* base = (k0 < PF)
            ? (S_poly  + (size_t)r * OPD2)
            : (S_graph + (size_t)(r - PF) * OPD2);
        v2f bv;
        bv.x = base[n];                 // row r
        bv.y = base[OPD2 + n];          // row r+1
        bfrag[s] = bv;
    }

    // Sweep all 16 M-tiles, reusing B fragments from registers.
    for (int mt = 0; mt < 16; ++mt) {
        v8f c = {};
        const float* arow = interm + (size_t)(mt * 16 + (lane & 15)) * KPAD + koff;
        #pragma unroll
        for (int s = 0; s < 15; ++s) {
            v2f a = *(const v2f*)(arow + 4 * s);    // 8B-aligned (koff even)
            c = __builtin_amdgcn_wmma_f32_16x16x4_f32(
                    /*neg_a=*/false, a, /*neg_b=*/false, bfrag[s],
                    /*c_mod=*/(short)0, c, /*reuse_a=*/false, /*reuse_b=*/false);
        }
        float* op = out + (size_t)(mt * 16 + 8 * (lane >> 4)) * OPD2 + n;
        #pragma unroll
        for (int j = 0; j < 8; ++j)
            op[(size_t)j * OPD2] = c[j];
    }
}

extern "C" void kernel_launch(void* const* d_in, const int* in_sizes, int n_in,
                              void* d_out, int out_size, void* d_ws, size_t ws_size,
                              hipStream_t stream) {
    // setup_inputs() order:
    const float* positions   = (const float*)d_in[0];   // (256, 2)
    const float* obs_pos     = (const float*)d_in[1];   // (2048, 2)
    const float* poly_dic    = (const float*)d_in[2];   // (2048, 28)
    const float* graph_dic   = (const float*)d_in[3];   // (2048, 512)
    const float* alpha_poly  = (const float*)d_in[4];   // (28, 28)
    const float* alpha_graph = (const float*)d_in[5];   // (512, 32)
    const float* S_poly      = (const float*)d_in[6];   // (28, 256, 256)
    const float* S_graph     = (const float*)d_in[7];   // (32, 256, 256)
    float* out = (float*)d_out;                         // (256, 256, 256)

    float* interm = (float*)d_ws;                       // 256*64 f32 = 64 KB

    build_interm_kernel<<<BATCH, 64, 0, stream>>>(
        positions, obs_pos, poly_dic, graph_dic, alpha_poly, alpha_graph, interm);

    // 4096 N-tiles, 8 waves (one tile each) per 256-thread block -> 512 blocks
    opd_wmma_gemm_kernel<<<512, 256, 0, stream>>>(interm, S_poly, S_graph, out);
}